// IWSA_65180423685345
// MI455X (gfx1250) — compile-verified
//
#include <hip/hip_runtime.h>
#include <hip/hip_bf16.h>

typedef float v2f __attribute__((ext_vector_type(2)));
typedef float v8f __attribute__((ext_vector_type(8)));

#define DIM   384
#define HEADS 12
#define HD    32
#define NTOK  3136      // 56*56
#define BATCH 16
#define IMGW  56
#define NWIN  64        // 8*8 windows
#define WS    7

// async global->LDS copy of 16 bytes (per-lane). VDST = LDS byte offset
// (low 32 bits of the generic LDS pointer ARE the wave-relative LDS address
// per the flat-aperture mapping), VADDR = 64-bit global address.
__device__ __forceinline__ void async_b128(const float* g, float* l)
{
    unsigned           loff = (unsigned)(unsigned long long)(void*)l;
    unsigned long long ga   = (unsigned long long)(const void*)g;
    asm volatile("global_load_async_to_lds_b128 %0, %1, off"
                 :: "v"(loff), "v"(ga) : "memory");
}
__device__ __forceinline__ void wait_async0()
{
    asm volatile("s_wait_asynccnt 0x0" ::: "memory");
}

#define WMMA_F32(A_, B_, C_) \
    __builtin_amdgcn_wmma_f32_16x16x4_f32(false, (A_), false, (B_), (short)0, (C_), false, false)

// ---------------------------------------------------------------------------
// fp32 WMMA GEMM:  C[M,Nt] = A[M,K] @ B[K,Nt] (+ bias[Nt])
// Block tile 128x64, K-step 32, 256 threads = 8 waves. Each wave owns a
// 2x2 group of 16x16 C tiles (full A/B fragment reuse: 4 ds loads -> 4 WMMA).
// Double-buffered LDS fed by async global->LDS B128 copies (ASYNCcnt).
// LDS pitches 36/68: multiples of 4 floats (16B-aligned B128 stores) and
// conflict-free for the strided A-fragment reads (gcd(36,64)=4 -> 16 banks).
// Requires M%128==0, Nt%64==0, K%32==0 (true for all calls here).
// ---------------------------------------------------------------------------
__global__ __launch_bounds__(256) void gemm_wmma_f32(
    const float* __restrict__ A, const float* __restrict__ Bm,
    const float* __restrict__ bias, float* __restrict__ C,
    int M, int Nt, int K)
{
    __shared__ float As[2][128][36];
    __shared__ float Bs[2][32][68];

    const int tid  = threadIdx.x;
    const int lane = tid & 31;
    const int lm   = lane & 15;    // M/N index within fragment
    const int lh   = lane >> 4;    // lane half selects K pair / row half
    const int wv   = tid >> 5;     // wave 0..7
    const int m0   = blockIdx.y * 128;
    const int n0   = blockIdx.x * 64;
    const int r0   = (wv >> 1) * 16;      // row tiles r0, r0+64
    const int r1   = r0 + 64;
    const int c0   = (wv & 1) * 16;       // col tiles c0, c0+32
    const int c1   = c0 + 32;

    v8f acc00 = {}, acc01 = {}, acc10 = {}, acc11 = {};

    // prefetch lambda-ish: issue async copies of tile at k0 into buffer bf
    auto prefetch = [&](int k0, int bf) {
        #pragma unroll
        for (int i = 0; i < 4; ++i) {               // A tile 128x32 = 1024 f4
            const int idx = tid + i * 256;
            const int row = idx >> 3, cf = (idx & 7) << 2;
            async_b128(&A[(size_t)(m0 + row) * K + k0 + cf], &As[bf][row][cf]);
        }
        #pragma unroll
        for (int i = 0; i < 2; ++i) {               // B tile 32x64 = 512 f4
            const int idx = tid + i * 256;
            const int row = idx >> 4, cf = (idx & 15) << 2;
            async_b128(&Bm[(size_t)(k0 + row) * Nt + n0 + cf], &Bs[bf][row][cf]);
        }
    };

    int buf = 0;
    prefetch(0, 0);
    wait_async0();
    __syncthreads();

    for (int k0 = 0; k0 < K; k0 += 32) {
        if (k0 + 32 < K) prefetch(k0 + 32, buf ^ 1);

        #pragma unroll
        for (int kk = 0; kk < 32; kk += 4) {
            const int kr = kk + 2 * lh;
            v2f a0; a0.x = As[buf][r0 + lm][kr]; a0.y = As[buf][r0 + lm][kr + 1];
            v2f a1; a1.x = As[buf][r1 + lm][kr]; a1.y = As[buf][r1 + lm][kr + 1];
            v2f b0; b0.x = Bs[buf][kr][c0 + lm]; b0.y = Bs[buf][kr + 1][c0 + lm];
            v2f b1; b1.x = Bs[buf][kr][c1 + lm]; b1.y = Bs[buf][kr + 1][c1 + lm];
            acc00 = WMMA_F32(a0, b0, acc00);
            acc01 = WMMA_F32(a0, b1, acc01);
            acc10 = WMMA_F32(a1, b0, acc10);
            acc11 = WMMA_F32(a1, b1, acc11);
        }

        wait_async0();      // prefetched tile (issued before compute) is ready
        __syncthreads();    // all waves done reading buf & writing buf^1
        buf ^= 1;
    }

    #pragma unroll
    for (int cc = 0; cc < 2; ++cc) {
        const int n  = n0 + (cc ? c1 : c0) + lm;
        const float bv = bias ? bias[n] : 0.0f;
        #pragma unroll
        for (int i = 0; i < 8; ++i) {
            const v8f& at = cc ? acc01 : acc00;
            const v8f& ab = cc ? acc11 : acc10;
            C[(size_t)(m0 + r0 + i + 8 * lh) * Nt + n] = at[i] + bv;
            C[(size_t)(m0 + r1 + i + 8 * lh) * Nt + n] = ab[i] + bv;
        }
    }
}

// ---------------------------------------------------------------------------
// Windowed attention: one block per (b, window, head). q/k/v 49x32 tiles are
// zero-padded to 64x32 in LDS so S = q k^T (64x64, K=32) and O = P v
// (64x32, K=64) run as pure 16x16x4 WMMA tilings. Softmax over the 49 valid
// columns; padded P columns are forced to zero so the O GEMM is exact.
// ---------------------------------------------------------------------------
__global__ __launch_bounds__(256) void attn_win(
    const float* __restrict__ qkv, float* __restrict__ sbuf)
{
    __shared__ float qs[64][33];
    __shared__ float ks[64][33];
    __shared__ float vs[64][33];
    __shared__ float ps[64][65];

    const int blk  = blockIdx.x;
    const int head = blk % HEADS;
    const int win  = (blk / HEADS) & (NWIN - 1);
    const int b    = blk / (HEADS * NWIN);
    const int wy   = win >> 3, wx = win & 7;
    const int tid  = threadIdx.x;

    // load q,k,v window tiles (rows >= 49 zero-padded)
    for (int i = tid; i < 64 * 32; i += 256) {
        const int row = i >> 5, col = i & 31;
        float vq = 0.f, vk = 0.f, vv = 0.f;
        if (row < WS * WS) {
            const int ty = row / WS, tx = row % WS;
            const int n  = (wy * WS + ty) * IMGW + (wx * WS + tx);
            const size_t base = ((size_t)(b * NTOK + n)) * (3 * DIM) + head * HD + col;
            vq = qkv[base];
            vk = qkv[base + DIM];
            vv = qkv[base + 2 * DIM];
        }
        qs[row][col] = vq; ks[row][col] = vk; vs[row][col] = vv;
    }
    __syncthreads();

    const int lane = tid & 31;
    const int lm   = lane & 15;
    const int lh   = lane >> 4;
    const int wv   = tid >> 5;
    const float scale = 0.17677669529663687f;   // 1/sqrt(32)

    // S = q @ k^T : 16 tiles, 8 waves -> 2 tiles each
    #pragma unroll
    for (int tt = 0; tt < 2; ++tt) {
        const int t  = wv + tt * 8;
        const int rB = (t >> 2) * 16, cBt = (t & 3) * 16;
        v8f acc = {};
        #pragma unroll
        for (int kk = 0; kk < 32; kk += 4) {
            const int kr = kk + 2 * lh;
            v2f a;  a.x  = qs[rB + lm][kr];  a.y  = qs[rB + lm][kr + 1];
            v2f bf; bf.x = ks[cBt + lm][kr]; bf.y = ks[cBt + lm][kr + 1];
            acc = WMMA_F32(a, bf, acc);
        }
        #pragma unroll
        for (int i = 0; i < 8; ++i)
            ps[rB + i + 8 * lh][cBt + lm] = acc[i] * scale;
    }
    __syncthreads();

    // softmax over valid 49 columns; zero the padded region of P
    for (int r = tid; r < 64; r += 256) {
        if (r < WS * WS) {
            float mx = -1e30f;
            for (int j = 0; j < WS * WS; ++j) mx = fmaxf(mx, ps[r][j]);
            float s = 0.f;
            for (int j = 0; j < WS * WS; ++j) {
                float e = __expf(ps[r][j] - mx);
                ps[r][j] = e; s += e;
            }
            const float inv = 1.0f / s;
            for (int j = 0; j < WS * WS; ++j) ps[r][j] *= inv;
            for (int j = WS * WS; j < 64; ++j) ps[r][j] = 0.f;
        } else {
            for (int j = 0; j < 64; ++j) ps[r][j] = 0.f;
        }
    }
    __syncthreads();

    // O = P @ V : 4x2 tiles, one per wave
    {
        const int rB = (wv >> 1) * 16, cBt = (wv & 1) * 16;
        v8f acc = {};
        #pragma unroll
        for (int kk = 0; kk < 64; kk += 4) {
            const int kr = kk + 2 * lh;
            v2f a;  a.x  = ps[rB + lm][kr];  a.y  = ps[rB + lm][kr + 1];
            v2f bf; bf.x = vs[kr][cBt + lm]; bf.y = vs[kr + 1][cBt + lm];
            acc = WMMA_F32(a, bf, acc);
        }
        #pragma unroll
        for (int i = 0; i < 8; ++i) {
            const int m = rB + i + 8 * lh;
            if (m < WS * WS) {
                const int ty = m / WS, tx = m % WS;
                const int n  = (wy * WS + ty) * IMGW + (wx * WS + tx);
                sbuf[((size_t)(b * NTOK + n)) * DIM + head * HD + cBt + lm] = acc[i];
            }
        }
    }
}

// ---------------------------------------------------------------------------
// Depthwise 3x3 SAME conv on v (v lives inside the qkv buffer at offset 2C),
// accumulated into sbuf (which attn_win fully wrote first). Coalesced over c.
// ---------------------------------------------------------------------------
__global__ __launch_bounds__(256) void dwconv_add(
    const float* __restrict__ qkv, const float* __restrict__ wconv,
    const float* __restrict__ bconv, float* __restrict__ sbuf)
{
    const int idx = blockIdx.x * 256 + threadIdx.x;
    const int total = BATCH * NTOK * DIM;
    if (idx >= total) return;
    const int c = idx % DIM;
    const int n = (idx / DIM) % NTOK;
    const int b = idx / (DIM * NTOK);
    const int h = n / IMGW, w = n % IMGW;

    float acc = bconv[c];
    #pragma unroll
    for (int ky = 0; ky < 3; ++ky) {
        const int hh = h + ky - 1;
        if (hh < 0 || hh >= IMGW) continue;
        #pragma unroll
        for (int kx = 0; kx < 3; ++kx) {
            const int ww = w + kx - 1;
            if (ww < 0 || ww >= IMGW) continue;
            const int n2 = hh * IMGW + ww;
            acc += wconv[(ky * 3 + kx) * DIM + c] *
                   qkv[((size_t)(b * NTOK + n2)) * (3 * DIM) + 2 * DIM + c];
        }
    }
    sbuf[idx] += acc;
}

// ---------------------------------------------------------------------------
extern "C" void kernel_launch(void* const* d_in, const int* in_sizes, int n_in,
                              void* d_out, int out_size, void* d_ws, size_t ws_size,
                              hipStream_t stream) {
    const float* x      = (const float*)d_in[0];
    const float* w_qkv  = (const float*)d_in[1];
    const float* w_proj = (const float*)d_in[2];
    const float* b_proj = (const float*)d_in[3];
    const float* w_conv = (const float*)d_in[4];
    const float* b_conv = (const float*)d_in[5];
    float* out = (float*)d_out;

    const int M = BATCH * NTOK;                      // 50176
    float* qkv  = (float*)d_ws;                      // M x 1152
    float* sbuf = (float*)((char*)d_ws + (size_t)M * 3 * DIM * sizeof(float));  // M x 384

    // 1) qkv = x @ w_qkv
    dim3 g1((3 * DIM) / 64, M / 128);
    gemm_wmma_f32<<<g1, 256, 0, stream>>>(x, w_qkv, nullptr, qkv, M, 3 * DIM, DIM);

    // 2) windowed attention -> sbuf
    attn_win<<<BATCH * NWIN * HEADS, 256, 0, stream>>>(qkv, sbuf);

    // 3) depthwise conv(v) + bias accumulated into sbuf
    dwconv_add<<<(BATCH * NTOK * DIM + 255) / 256, 256, 0, stream>>>(
        qkv, w_conv, b_conv, sbuf);

    // 4) out = sbuf @ w_proj + b_proj
    dim3 g2(DIM / 64, M / 128);
    gemm_wmma_f32<<<g2, 256, 0, stream>>>(sbuf, w_proj, b_proj, out, M, DIM, DIM);
}